// LocalizationVAE1_15539191677794
// MI455X (gfx1250) — compile-verified
//
#include <hip/hip_runtime.h>

typedef float v2f __attribute__((ext_vector_type(2)));
typedef float v8f __attribute__((ext_vector_type(8)));
typedef unsigned int u32x4 __attribute__((ext_vector_type(4)));
typedef int i32x4 __attribute__((ext_vector_type(4)));
typedef int i32x8 __attribute__((ext_vector_type(8)));

#define NXc 64
#define NYc 64
#define Pc 6
#define PHW 3
#define I0c 1000.0f
// 1/(sqrt(2)*0.92) = 1/1.30107647738
#define INV_ALPHA 0.76859444f

// Branchless erf, Abramowitz-Stegun 7.1.26, |abs err| <= 1.5e-7.
__device__ __forceinline__ float erf_fast(float x) {
  float ax = __builtin_fabsf(x);
  float t = __builtin_amdgcn_rcpf(__builtin_fmaf(0.3275911f, ax, 1.0f));
  float p = t * (0.254829592f +
            t * (-0.284496736f +
            t * (1.421413741f +
            t * (-1.453152027f +
            t * 1.061405429f))));
  float e = __expf(-ax * ax);
  float r = __builtin_fmaf(-p, e, 1.0f);
  return __builtin_copysignf(r, x);
}

__global__ __launch_bounds__(256) void spot_render_kernel(
    const float* __restrict__ z, float* __restrict__ out) {
  __shared__ float tile[NXc * NYc];  // 16 KB per-image accumulator

  const int tid = threadIdx.x;
  const int b = blockIdx.x;

  // ---- zero the LDS tile ----
  float4* t4 = (float4*)tile;
  const float4 z4 = make_float4(0.f, 0.f, 0.f, 0.f);
#pragma unroll
  for (int r = 0; r < 4; ++r) t4[r * 256 + tid] = z4;
  __syncthreads();

  const int lane = tid & 31;
  const int wave = tid >> 5;
  const float* zb = z + (size_t)b * (2 * 64);

  // Lane groups of 8 evaluate erf boundary values E[t] = erf((t-0.5-t0p)/alpha):
  //   lanes 0-7  : t0p = x0p of spot0   (Ex0[t], t = lane)
  //   lanes 8-15 : t0p = y0p of spot0   (Ey0[t])
  //   lanes 16-23: t0p = x0p of spot1   (Ex1[t])
  //   lanes 24-31: t0p = y0p of spot1   (Ey1[t])
  const float c = (float)(lane & 7) - 0.5f;
  const int grp = lane >> 3;

  // ---- 8 waves x 4 pairs = 64 spots; 2 spots per WMMA (block-diagonal K packing) ----
#pragma unroll
  for (int k = 0; k < 4; ++k) {
    const int s0 = wave * 8 + 2 * k;
    const float x00 = zb[s0],     y00 = zb[64 + s0];
    const float x01 = zb[s0 + 1], y01 = zb[64 + s0 + 1];
    const int px0 = (int)rintf(x00) - PHW, py0 = (int)rintf(y00) - PHW;
    const int px1 = (int)rintf(x01) - PHW, py1 = (int)rintf(y01) - PHW;
    const bool valid0 = (px0 >= 0) && (px0 < NXc - Pc) && (py0 >= 0) && (py0 < NYc - Pc);
    const bool valid1 = (px1 >= 0) && (px1 < NXc - Pc) && (py1 >= 0) && (py1 < NYc - Pc);
    const float x0p0 = x00 - (float)px0, y0p0 = y00 - (float)py0;
    const float x0p1 = x01 - (float)px1, y0p1 = y01 - (float)py1;

    const float sel = (grp == 0) ? x0p0 : (grp == 1) ? y0p0 : (grp == 2) ? x0p1 : y0p1;
    const float E  = erf_fast((c - sel) * INV_ALPHA);
    const float S0 = __shfl(E, lane + 1);   // E[t+1] within own group
    const float S1 = __shfl(E, lane + 8);   // next group's E[t]
    const float S2 = __shfl(E, lane + 9);   // next group's E[t+1]
    const float dA = 0.5f * (S0 - E);       // lanes 0-5: lx0[t]   | lanes 24-29: ly1[t]
    const float dB = 0.5f * (S2 - S1);      // lanes 0-5: ly0[t]   | lanes 8-13:  lx1[t]

    const bool low6 = (lane < 6);
    const bool mid6 = (lane >= 8) && (lane < 14);
    const bool hi6  = (lane >= 24) && (lane < 30);

    // A (16x4 f32): K=0 col = I0*lx0 (rows 0-5), K=1 col = I0*lx1 (rows 8-13).
    // B (4x16 f32): K=0 row = ly0 (cols 0-5, lanes 0-5),
    //               K=1 row = ly1 (cols 8-13, lanes 24-29). K=2,3 zero.
    v2f A;  A.x  = low6 ? I0c * dA : 0.f;  A.y  = mid6 ? I0c * dB : 0.f;
    v2f Bm; Bm.x = low6 ? dB : (hi6 ? dA : 0.f);  Bm.y = 0.f;
    v8f C = {};
    // D = lx0 (x) ly0 in quad(0-5,0-5) + lx1 (x) ly1 in quad(8-13,8-13); cross terms 0.
    v8f d = __builtin_amdgcn_wmma_f32_16x16x4_f32(
        false, A, false, Bm, (short)0, C, false, false);

    // Scatter: lanes 0-5 own spot0 cols (N=lane, M=i via VGPR i);
    //          lanes 24-29 own spot1 cols (N=8+n, M=8+i via VGPR i).
    const bool act = (low6 && valid0) || (hi6 && valid1);
    if (act) {
      const int px = low6 ? px0 : px1;
      const int py = low6 ? py0 : py1;
      const int j  = low6 ? lane : lane - 24;
      const int base = px * NYc + py + j;
#pragma unroll
      for (int i = 0; i < Pc; ++i) {
        atomicAdd(&tile[base + i * NYc], d[i]);  // ds_add_f32
      }
    }
  }
  __syncthreads();

#if __has_builtin(__builtin_amdgcn_tensor_store_from_lds) && \
    __has_builtin(__builtin_amdgcn_s_wait_tensorcnt)
  // ---- writeback: one TDM DMA moves the whole 16 KB tile LDS -> HBM ----
  if (tid == 0) {
    const unsigned long long gaddr =
        (unsigned long long)(uintptr_t)(out + (size_t)b * (NXc * NYc));
    const unsigned int laddr = (unsigned int)(uintptr_t)tile;  // flat[31:0] = LDS byte offset

    // D# group 0: count=1 | lds_addr | global_addr[56:0] | type=2
    u32x4 g0;
    g0.x = 1u;
    g0.y = laddr;
    g0.z = (unsigned int)gaddr;
    g0.w = (unsigned int)((gaddr >> 32) & 0x01FFFFFFu) | 0x80000000u;

    // D# group 1: data_size=4B; tensor_dim0=4096, tensor_dim1=1;
    // tile_dim0=4096, tile_dim1=1 (single row); tensor_dim0_stride=4096.
    i32x8 g1;
    g1[0] = (int)(2u << 16);        // workgroup_mask=0, data_size=2 (4 bytes)
    g1[1] = (int)(4096u << 16);     // tensor_dim0[15:0] in bits[63:48]
    g1[2] = (int)(1u << 16);        // tensor_dim0[31:16]=0 | tensor_dim1[15:0]=1
    g1[3] = (int)(4096u << 16);     // tensor_dim1[31:16]=0 | tile_dim0=4096
    g1[4] = 1;                      // tile_dim1=1, tile_dim2=0
    g1[5] = 4096;                   // tensor_dim0_stride[31:0]
    g1[6] = (int)(4096u << 16);     // stride0[47:32]=0 | tensor_dim1_stride[15:0]=4096
    g1[7] = 0;                      // tensor_dim1_stride[47:16]=0

    // D# group 2: tensor_dim2=1, tensor_dim3=1, stride2=0, tile_dim3=1
    i32x4 g2; g2[0] = 1; g2[1] = 1; g2[2] = 0; g2[3] = (int)(1u << 16);
    // D# group 3: stride3=0, tensor_dim4=1, tile_dim4=1
    i32x4 g3; g3[0] = 0; g3[1] = (int)(1u << 16); g3[2] = (int)(1u << 16); g3[3] = 0;

    // 6-arg toolchain form (clang-23 / therock-10.0): extra 256-bit operand,
    // zero-filled (matches the probe-verified zero-filled call).
    i32x8 gx = {0, 0, 0, 0, 0, 0, 0, 0};

    __builtin_amdgcn_tensor_store_from_lds(g0, g1, g2, g3, gx, 0);
    __builtin_amdgcn_s_wait_tensorcnt(0);
  }
#else
  // ---- fallback: coalesced b128 writeback ----
  float* ob = out + (size_t)b * (NXc * NYc);
  float4* o4 = (float4*)ob;
#pragma unroll
  for (int r = 0; r < 4; ++r) o4[r * 256 + tid] = t4[r * 256 + tid];
#endif
}

extern "C" void kernel_launch(void* const* d_in, const int* in_sizes, int n_in,
                              void* d_out, int out_size, void* d_ws,
                              size_t ws_size, hipStream_t stream) {
  (void)n_in; (void)d_ws; (void)ws_size; (void)out_size;
  const float* z = (const float*)d_in[0];
  float* out = (float*)d_out;
  const int B = in_sizes[0] / (2 * 64);  // 8192
  spot_render_kernel<<<B, 256, 0, stream>>>(z, out);
}